// CPGNetwork_88347477278757
// MI455X (gfx1250) — compile-verified
//
#include <hip/hip_runtime.h>
#include <hip/hip_bf16.h>

// ---------------------------------------------------------------------------
// CPG network: in-MLP (GEMV chain) -> RK4(3/8) CPG ODE -> out-MLP (GEMV chain)
// All fp32. Bandwidth-bound (0.5 flop/byte, ~161 MB weights -> ~6.9us floor
// at 23.3 TB/s). GEMV tiles use V_WMMA_F32_16X16X4_F32: one WMMA consumes a
// 16x4 fp32 tile of W (the mandatory HBM traffic) with the x-chunk in column
// 0 of B; split-K across the 8 waves of a block + tiny LDS reduction.
// ---------------------------------------------------------------------------

typedef __attribute__((ext_vector_type(2))) float v2f;
typedef __attribute__((ext_vector_type(8))) float v8f;

// y[M] = act(W[M,K] @ x[K] + b[M]);  one block per 16 rows, 8 waves split K.
__global__ __launch_bounds__(256) void gemv_wmma_f32(
    const float* __restrict__ W, const float* __restrict__ bias,
    const float* __restrict__ x, float* __restrict__ y,
    int M, int K, int relu)
{
    __shared__ float red[8][16];

    const int lane = threadIdx.x & 31;
    const int wave = threadIdx.x >> 5;
    const int row0 = blockIdx.x << 4;          // 16 rows per block (M % 16 == 0)
    const int lm   = lane & 15;                // row within tile (A), col (B/C)
    const int kh   = lane >> 4;                // K half-pair selector: 0 -> k,k+1 ; 1 -> k+2,k+3

    const float* __restrict__ wrow = W + (size_t)(row0 + lm) * (size_t)K;

    // split K over 8 waves, chunk rounded to 16 so the main loop is clean
    const int Kc   = (((K + 7) >> 3) + 15) & ~15;
    const int kbeg = wave * Kc;
    const int kend = min(kbeg + Kc, K);

    v8f acc = {};

    int k = kbeg;
    // main loop: 16 K-values (4 WMMAs) per iteration, b64 A-loads per lane
    for (; k + 16 <= kend; k += 16) {
        __builtin_prefetch(&wrow[k + 64], 0, 1);   // global_prefetch_b8, ~4 iters ahead
#pragma unroll
        for (int u = 0; u < 4; ++u) {
            const int kb = k + u * 4 + kh * 2;
            v2f a, b;
            a.x = wrow[kb];
            a.y = wrow[kb + 1];
            b.x = (lm == 0) ? x[kb]     : 0.0f;    // vector in column N=0 of B
            b.y = (lm == 0) ? x[kb + 1] : 0.0f;
            acc = __builtin_amdgcn_wmma_f32_16x16x4_f32(
                false, a, false, b, (short)0, acc, false, false);
        }
    }
    // predicated tail (covers the K=2049 case)
    for (; k < kend; k += 4) {
        const int kb = k + kh * 2;
        v2f a, b;
        a.x = (kb     < kend) ? wrow[kb]     : 0.0f;
        a.y = (kb + 1 < kend) ? wrow[kb + 1] : 0.0f;
        b.x = (lm == 0 && kb     < kend) ? x[kb]     : 0.0f;
        b.y = (lm == 0 && kb + 1 < kend) ? x[kb + 1] : 0.0f;
        acc = __builtin_amdgcn_wmma_f32_16x16x4_f32(
            false, a, false, b, (short)0, acc, false, false);
    }

    // column N=0 of C: lane 0 holds rows 0..7 in acc[0..7], lane 16 rows 8..15
    if (lm == 0) {
#pragma unroll
        for (int v = 0; v < 8; ++v)
            red[wave][kh * 8 + v] = acc[v];
    }
    __syncthreads();

    if (threadIdx.x < 16) {
        const int m = row0 + threadIdx.x;
        float s = bias[m];
#pragma unroll
        for (int w = 0; w < 8; ++w)
            s += red[w][threadIdx.x];
        if (relu) s = fmaxf(s, 0.0f);
        y[m] = s;
    }
}

// xin[0..2047] = x, xin[2048] = timestep
__global__ void build_input_k(const float* __restrict__ x,
                              const float* __restrict__ ts,
                              float* __restrict__ xin)
{
    const int i = blockIdx.x * blockDim.x + threadIdx.x;
    if (i < 2048)       xin[i] = x[i];
    else if (i == 2048) xin[i] = *ts;
}

// ---- CPG ODE, RK4 3/8 rule, one wave32, lane i = oscillator i -------------
__device__ inline void cpg_deriv(float a, float d, float p,
                                 float ia_i, float ifr_i,
                                 const float* __restrict__ cwrow,
                                 const float* __restrict__ pbrow,
                                 float& da, float& dd, float& dp)
{
    float s = 0.0f;
#pragma unroll 8
    for (int j = 0; j < 32; ++j) {
        const float pj = __shfl(p, j, 32);
        s += cwrow[j] * __sinf(pj - p - pbrow[j]);
    }
    // derivative slots exactly as in the reference concatenation:
    da = d;                                        // d(slot0)/dt = a_dot
    dd = ifr_i + a * s;                            // d(slot1)/dt = phase_dots
    dp = 1000.0f * (250.0f * (ia_i - a) - d);      // d(slot2)/dt = a_dd
}

__global__ __launch_bounds__(32) void cpg_rk4_k(
    const float* __restrict__ state, const float* __restrict__ params,
    const float* __restrict__ ts, float* __restrict__ new_state,
    float* __restrict__ cpg_out)
{
    const int i = threadIdx.x;                // 0..31
    const float h = *ts;
    const float ia_i  = params[i];
    const float ifr_i = params[32 + i];
    const float* cwrow = params + 64        + i * 32;
    const float* pbrow = params + 64 + 1024 + i * 32;

    const float ya = state[i], yd = state[32 + i], yp = state[64 + i];

    float k1a, k1d, k1p, k2a, k2d, k2p, k3a, k3d, k3p, k4a, k4d, k4p;
    cpg_deriv(ya, yd, yp, ia_i, ifr_i, cwrow, pbrow, k1a, k1d, k1p);
    cpg_deriv(ya + h * k1a / 3.0f, yd + h * k1d / 3.0f, yp + h * k1p / 3.0f,
              ia_i, ifr_i, cwrow, pbrow, k2a, k2d, k2p);
    cpg_deriv(ya + h * (k2a - k1a / 3.0f), yd + h * (k2d - k1d / 3.0f),
              yp + h * (k2p - k1p / 3.0f),
              ia_i, ifr_i, cwrow, pbrow, k3a, k3d, k3p);
    cpg_deriv(ya + h * (k1a - k2a + k3a), yd + h * (k1d - k2d + k3d),
              yp + h * (k1p - k2p + k3p),
              ia_i, ifr_i, cwrow, pbrow, k4a, k4d, k4p);

    const float na = ya + h * 0.125f * (k1a + 3.0f * (k2a + k3a) + k4a);
    const float nd = yd + h * 0.125f * (k1d + 3.0f * (k2d + k3d) + k4d);
    const float np = yp + h * 0.125f * (k1p + 3.0f * (k2p + k3p) + k4p);

    new_state[i]      = na;
    new_state[32 + i] = nd;
    new_state[64 + i] = np;

    // reference: a = new_state[:n]; ph = new_state[n:2n]
    const float A = na, P = nd;
    cpg_out[i]      = A * __cosf(P);
    cpg_out[32 + i] = A * __sinf(P);
}

// ---------------------------------------------------------------------------

extern "C" void kernel_launch(void* const* d_in, const int* in_sizes, int n_in,
                              void* d_out, int out_size, void* d_ws, size_t ws_size,
                              hipStream_t stream)
{
    const float* state = (const float*)d_in[0];   // 96
    const float* x     = (const float*)d_in[1];   // 2048
    const float* ts    = (const float*)d_in[2];   // 1
    const float* inW0  = (const float*)d_in[3];   // 4096x2049
    const float* inb0  = (const float*)d_in[4];
    const float* inW1  = (const float*)d_in[5];   // 4096x4096
    const float* inb1  = (const float*)d_in[6];
    const float* inW2  = (const float*)d_in[7];   // 2112x4096
    const float* inb2  = (const float*)d_in[8];
    const float* oW0   = (const float*)d_in[9];   // 2048x64
    const float* ob0   = (const float*)d_in[10];
    const float* oW1   = (const float*)d_in[11];  // 2048x2048
    const float* ob1   = (const float*)d_in[12];
    const float* oW2   = (const float*)d_in[13];  // 1024x2048
    const float* ob2   = (const float*)d_in[14];

    float* out = (float*)d_out;                   // [new_state(96) | out(1024)]

    float* ws     = (float*)d_ws;
    float* xin    = ws;            // 2049  (pad to 2064)
    float* h1     = ws + 2064;     // 4096
    float* h2     = ws + 6160;     // 4096
    float* params = ws + 10256;    // 2112
    float* cpg    = ws + 12368;    // 64
    float* o1     = ws + 12432;    // 2048
    float* o2     = ws + 14480;    // 2048   total ~66 KB << ws_size

    build_input_k<<<(2049 + 255) / 256, 256, 0, stream>>>(x, ts, xin);

    // in-MLP
    gemv_wmma_f32<<<4096 / 16, 256, 0, stream>>>(inW0, inb0, xin, h1, 4096, 2049, 1);
    gemv_wmma_f32<<<4096 / 16, 256, 0, stream>>>(inW1, inb1, h1, h2, 4096, 4096, 1);
    gemv_wmma_f32<<<2112 / 16, 256, 0, stream>>>(inW2, inb2, h2, params, 2112, 4096, 0);

    // CPG RK4 step (writes new_state to d_out[0:96], cpg_out(64) to ws)
    cpg_rk4_k<<<1, 32, 0, stream>>>(state, params, ts, out, cpg);

    // out-MLP
    gemv_wmma_f32<<<2048 / 16, 256, 0, stream>>>(oW0, ob0, cpg, o1, 2048, 64, 1);
    gemv_wmma_f32<<<2048 / 16, 256, 0, stream>>>(oW1, ob1, o1, o2, 2048, 2048, 1);
    gemv_wmma_f32<<<1024 / 16, 256, 0, stream>>>(oW2, ob2, o2, out + 96, 1024, 2048, 0);
}